// a3c_lstm_ga_attention_multigru_50929722196529
// MI455X (gfx1250) — compile-verified
//
#include <hip/hip_runtime.h>
#include <hip/hip_bf16.h>
#include <math.h>

// ---------------------------------------------------------------------------
// a3c_lstm_ga_attention_multigru — CDNA5 (gfx1250) implementation.
// fp32 throughout (matches reference). The batched GEMM (GRU input projection)
// uses native fp32 WMMA (V_WMMA_F32_16X16X4_F32), 1 M-tile x 4 N-tiles per
// wave with A-fragment reuse. Sequential GRU recurrence is latency-bound ->
// 6 concurrent workgroups, h/gh in LDS, float4 weight streaming from L2.
// ---------------------------------------------------------------------------

typedef __attribute__((ext_vector_type(2))) float v2f;
typedef __attribute__((ext_vector_type(8))) float v8f;

#if defined(__has_builtin)
#if __has_builtin(__builtin_amdgcn_wmma_f32_16x16x4_f32)
#define HAVE_WMMA_F32X4 1
#endif
#endif

#define H 256
#define E 32
#define IMG_FLAT 2304   // 64*6*6
#define P_LEN 512
#define C_LEN 256
#define N_LEN 512
#define TOT_ROWS 1280   // P+C+N

// Workspace layout (float offsets)
#define X1_OFF    0u            // 128*30*30 = 115200
#define X2_OFF    115200u       // 64*14*14  = 12544
#define XIMG_OFF  127744u       // 2304
#define EALL_OFF  130048u       // 1280*32   = 40960
#define GI_OFF    171008u       // 2560*768  = 1966080
#define HF_OFF    2137088u      // 1280*256  = 327680
#define HB_OFF    2464768u      // 1280*256  = 327680
#define AKIN_OFF  2792448u      // 768
#define PSUM_OFF  2793216u      // 512
#define NSUM_OFF  2793728u      // 512

__device__ __forceinline__ float sigf(float x) { return 1.0f / (1.0f + expf(-x)); }

// ---------------------------------------------------------------------------
// Direct conv (VALID, stride S) + ReLU. One thread per output element.
// ---------------------------------------------------------------------------
__global__ void k_conv_relu(const float* __restrict__ in, const float* __restrict__ w,
                            const float* __restrict__ bias, float* __restrict__ out,
                            int IC, int IH, int IW, int OC, int OH, int OW, int K, int S) {
    int idx = blockIdx.x * blockDim.x + threadIdx.x;
    int total = OC * OH * OW;
    if (idx >= total) return;
    int x = idx % OW;
    int y = (idx / OW) % OH;
    int o = idx / (OW * OH);
    float acc = bias[o];
    const float* wo = w + (size_t)o * IC * K * K;
    for (int c = 0; c < IC; ++c) {
        const float* ip = in + ((size_t)c * IH + y * S) * IW + x * S;
        const float* wp = wo + (size_t)c * K * K;
        for (int ky = 0; ky < K; ++ky)
            for (int kx = 0; kx < K; ++kx)
                acc = fmaf(ip[ky * IW + kx], wp[ky * K + kx], acc);
    }
    out[idx] = fmaxf(acc, 0.0f);
}

// ---------------------------------------------------------------------------
// Embedding gather: rows [0,512)=prev, [512,768)=curr, [768,1280)=next.
// ---------------------------------------------------------------------------
__global__ void k_embed(const int* __restrict__ prev, const int* __restrict__ curr,
                        const int* __restrict__ nxt, const float* __restrict__ emb,
                        float* __restrict__ e_all) {
    int idx = blockIdx.x * blockDim.x + threadIdx.x;
    if (idx >= TOT_ROWS * E) return;
    int row = idx >> 5, k = idx & 31;
    int tok;
    if (row < P_LEN)              tok = prev[row];
    else if (row < P_LEN + C_LEN) tok = curr[row - P_LEN];
    else                          tok = nxt[row - P_LEN - C_LEN];
    e_all[idx] = emb[(size_t)tok * E + k];
}

// ---------------------------------------------------------------------------
// GRU input projection: gi_j = E_seq @ wih_j^T + bih_j   (fp32 WMMA 16x16x4)
// One wave per (16-row M tile) x (4 x 16-col N tiles): the A fragment is
// loaded once per K-step and reused by 4 WMMAs. EXEC all ones (blockDim=32).
// A frag (16x4): lanes 0-15 carry rows M=lane, {K,K+1}; lanes 16-31 {K+2,K+3}.
// B frag (4x16): same K split, N=lane&15.  C/D: VGPR r -> M=r+8*(lane>=16).
// ---------------------------------------------------------------------------
#define NQ 4   // N tiles per wave
__global__ void k_gi_wmma(const float* __restrict__ e_all, const float* __restrict__ wih,
                          const float* __restrict__ bih, float* __restrict__ gi) {
    const int lensT[6]  = {512, 256, 512, 512, 256, 512};
    const int segoff[3] = {0, 512, 768};
    const int runoff[6] = {0, 512, 768, 1280, 1792, 2048};
    const int NGRP = 48 / NQ;   // 12 N-groups of 4 tiles

    int blk = blockIdx.x;
    int j = 0, rem = blk;
    for (; j < 6; ++j) {
        int nt = (lensT[j] / 16) * NGRP;
        if (rem < nt) break;
        rem -= nt;
    }
    if (j >= 6) return;
    int mt = rem / NGRP, ng = rem % NGRP;

    int lane = threadIdx.x;
    int half = lane >> 4;     // 0: lanes 0-15, 1: lanes 16-31
    int l15  = lane & 15;
    int mbase  = mt * 16;
    int nbase0 = ng * (16 * NQ);

    const float* Ee = e_all + (size_t)segoff[j % 3] * E;
    const float* W  = wih + (size_t)j * 768 * E;
    int arow = mbase + l15;   // A row (M) owned by this lane
    int koff = half * 2;

    v8f acc[NQ];
    for (int q = 0; q < NQ; ++q) acc[q] = (v8f){};

#ifdef HAVE_WMMA_F32X4
    for (int k0 = 0; k0 < E; k0 += 4) {
        v2f a;
        a.x = Ee[(size_t)arow * E + k0 + koff];
        a.y = Ee[(size_t)arow * E + k0 + koff + 1];
#pragma unroll
        for (int q = 0; q < NQ; ++q) {
            int bcol = nbase0 + q * 16 + l15;       // B[k][n] = wih[n][k]
            v2f b;
            b.x = W[(size_t)bcol * E + k0 + koff];
            b.y = W[(size_t)bcol * E + k0 + koff + 1];
            acc[q] = __builtin_amdgcn_wmma_f32_16x16x4_f32(
                /*neg_a=*/false, a, /*neg_b=*/false, b,
                /*c_mod=*/(short)0, acc[q], /*reuse_a=*/false, /*reuse_b=*/false);
        }
    }
#else
    // Scalar fallback with identical C ownership (lane -> M=r+8*half, N=l15).
    for (int q = 0; q < NQ; ++q) {
        int bcol = nbase0 + q * 16 + l15;
        for (int r = 0; r < 8; ++r) {
            float s = 0.0f;
            int m = mbase + r + half * 8;
            for (int k = 0; k < E; ++k)
                s = fmaf(Ee[(size_t)m * E + k], W[(size_t)bcol * E + k], s);
            acc[q][r] = s;
        }
    }
#endif
    float* Gq = gi + (size_t)runoff[j] * 768;
#pragma unroll
    for (int q = 0; q < NQ; ++q) {
        int ncol = nbase0 + q * 16 + l15;
        const float bi = bih[(size_t)j * 768 + ncol];
#pragma unroll
        for (int r = 0; r < 8; ++r) {
            int row = mbase + r + half * 8;
            Gq[(size_t)row * 768 + ncol] = acc[q][r] + bi;
        }
    }
}

// ---------------------------------------------------------------------------
// GRU recurrence. blockIdx.x = run j (0..5), 768 threads (24 waves).
// Per step: gh[u] = bhh[u] + h . whh[u]; then 256 gate lanes update h.
// Weight rows streamed as float4 (global_load_b128), h read as float4 from
// LDS (ds_load_b128) -> minimal issue slots on the serial dependence chain.
// ---------------------------------------------------------------------------
__global__ void k_gru(const float* __restrict__ gi, const float* __restrict__ whh,
                      const float* __restrict__ bhh,
                      float* __restrict__ hf, float* __restrict__ hb) {
    const int lensT[6]  = {512, 256, 512, 512, 256, 512};
    const int segoff[3] = {0, 512, 768};
    const int runoff[6] = {0, 512, 768, 1280, 1792, 2048};

    int j = blockIdx.x;
    int u = threadIdx.x;          // 0..767
    __shared__ __align__(16) float h[H];
    __shared__ float gh[3 * H];
    if (u < H) h[u] = 0.0f;
    __syncthreads();

    const int T = lensT[j];
    const float* G = gi + (size_t)runoff[j] * 768;
    const float4* wrow4 =
        reinterpret_cast<const float4*>(whh + (size_t)j * 768 * H + (size_t)u * H);
    const float4* h4p = reinterpret_cast<const float4*>(h);
    const float bh = bhh[(size_t)j * 768 + u];
    float* out = (j < 3 ? hf : hb) + (size_t)segoff[j % 3] * H;

    for (int t = 0; t < T; ++t) {
        float acc = bh;
#pragma unroll 8
        for (int k4 = 0; k4 < H / 4; ++k4) {
            float4 w4 = wrow4[k4];
            float4 hv = h4p[k4];
            acc = fmaf(hv.x, w4.x, acc);
            acc = fmaf(hv.y, w4.y, acc);
            acc = fmaf(hv.z, w4.z, acc);
            acc = fmaf(hv.w, w4.w, acc);
        }
        gh[u] = acc;
        __syncthreads();
        if (u < H) {
            const float* gr = G + (size_t)t * 768;
            float r  = sigf(gr[u] + gh[u]);
            float zg = sigf(gr[H + u] + gh[H + u]);
            float n  = tanhf(gr[2 * H + u] + r * gh[2 * H + u]);
            float h2 = (1.0f - zg) * n + zg * h[u];
            out[(size_t)t * H + u] = h2;
            h[u] = h2;
        }
        __syncthreads();
    }
}

// ---------------------------------------------------------------------------
// curr_rep + image_emb -> ak_in[768]. 768 threads.
// ak_in[0:256]=hf[767], [256:512]=hb[767], [512:768]=x_img.flat @ img_w^T + b
// ---------------------------------------------------------------------------
__global__ void k_curr(const float* __restrict__ hf, const float* __restrict__ hb,
                       const float* __restrict__ ximg, const float* __restrict__ img_w,
                       const float* __restrict__ img_b, float* __restrict__ akin) {
    int t = threadIdx.x;
    if (t < 256) {
        akin[t] = hf[(size_t)767 * H + t];
    } else if (t < 512) {
        akin[t] = hb[(size_t)767 * H + (t - 256)];
    } else {
        int o = t - 512;
        float a = img_b[o];
        const float4* w4 = reinterpret_cast<const float4*>(img_w + (size_t)o * IMG_FLAT);
        const float4* x4 = reinterpret_cast<const float4*>(ximg);
        for (int k = 0; k < IMG_FLAT / 4; ++k) {
            float4 wv = w4[k], xv = x4[k];
            a = fmaf(xv.x, wv.x, fmaf(xv.y, wv.y, fmaf(xv.z, wv.z, fmaf(xv.w, wv.w, a))));
        }
        akin[t] = a;
    }
}

// ---------------------------------------------------------------------------
// Bilinear attention. block 0 = prev, block 1 = next. 512 threads.
//   k  = ak_in @ akw^T + akb          (256)
//   kb = k @ bw                       (512)
//   s_i = kb . rep_i + bb ; softmax ; out = sum w_i rep_i   (512)
// rep_i rows: prev -> hf[i] | hb[1279-i];  next -> hf[768+i] | hb[511-i]
// ---------------------------------------------------------------------------
__global__ void k_attend(const float* __restrict__ akin,
                         const float* __restrict__ akp_w, const float* __restrict__ akp_b,
                         const float* __restrict__ akn_w, const float* __restrict__ akn_b,
                         const float* __restrict__ bpw, const float* __restrict__ bpb,
                         const float* __restrict__ bnw, const float* __restrict__ bnb,
                         const float* __restrict__ hf, const float* __restrict__ hb,
                         float* __restrict__ psum, float* __restrict__ nsum) {
    int b = blockIdx.x, t = threadIdx.x;   // 512 threads
    const float* akw = b ? akn_w : akp_w;
    const float* akb = b ? akn_b : akp_b;
    const float* bw  = b ? bnw : bpw;
    const float bb   = (b ? bnb : bpb)[0];
    float* outp = b ? nsum : psum;

    __shared__ __align__(16) float lk[256];
    __shared__ __align__(16) float lkb[512];
    __shared__ float wgt[512];
    __shared__ float red[512];

    if (t < 256) {
        float a = akb[t];
        const float* wr = akw + (size_t)t * 768;
        for (int k = 0; k < 768; ++k) a = fmaf(akin[k], wr[k], a);
        lk[t] = a;
    }
    __syncthreads();

    {
        float a = 0.0f;
        for (int m = 0; m < 256; ++m) a = fmaf(lk[m], bw[(size_t)m * 512 + t], a);
        lkb[t] = a;
    }
    __syncthreads();

    {
        int rf = b ? 768 + t : t;
        int rb = b ? 511 - t : 1279 - t;
        const float* hfr = hf + (size_t)rf * H;
        const float* hbr = hb + (size_t)rb * H;
        float s = bb;
        for (int m = 0; m < 256; ++m) s = fmaf(lkb[m], hfr[m], s);
        for (int m = 0; m < 256; ++m) s = fmaf(lkb[256 + m], hbr[m], s);
        wgt[t] = s;
        red[t] = s;
    }
    __syncthreads();

    for (int st = 256; st > 0; st >>= 1) {
        if (t < st) red[t] = fmaxf(red[t], red[t + st]);
        __syncthreads();
    }
    float mx = red[0];
    __syncthreads();
    float ev = expf(wgt[t] - mx);
    red[t] = ev;
    __syncthreads();
    for (int st = 256; st > 0; st >>= 1) {
        if (t < st) red[t] += red[t + st];
        __syncthreads();
    }
    float tot = red[0];
    __syncthreads();
    wgt[t] = ev / tot;
    __syncthreads();

    {
        int d = t;
        float acc = 0.0f;
        for (int i = 0; i < 512; ++i) {
            int rfi = b ? 768 + i : i;
            int rbi = b ? 511 - i : 1279 - i;
            float v = (d < 256) ? hf[(size_t)rfi * H + d]
                                : hb[(size_t)rbi * H + (d - 256)];
            acc = fmaf(wgt[i], v, acc);
        }
        outp[d] = acc;
    }
}

// ---------------------------------------------------------------------------
// Fused tail: pred/nred -> gate -> xg -> lin -> LSTM -> critic/actor.
// One block, 1024 threads. d_out = [crit(1), act(4), hx2(256), cx2(256)].
// ---------------------------------------------------------------------------
__global__ void k_tail(const float* __restrict__ akin,
                       const float* __restrict__ psum, const float* __restrict__ nsum,
                       const float* __restrict__ pred_w, const float* __restrict__ pred_b,
                       const float* __restrict__ nred_w, const float* __restrict__ nred_b,
                       const float* __restrict__ gate_w, const float* __restrict__ gate_b,
                       const float* __restrict__ ximg,
                       const float* __restrict__ lin_w, const float* __restrict__ lin_b,
                       const float* __restrict__ lstm_wih, const float* __restrict__ lstm_whh,
                       const float* __restrict__ lstm_bih, const float* __restrict__ lstm_bhh,
                       const float* __restrict__ hx, const float* __restrict__ cx,
                       const float* __restrict__ time_emb, const int* __restrict__ tx,
                       const float* __restrict__ crit_w, const float* __restrict__ crit_b,
                       const float* __restrict__ act_w, const float* __restrict__ act_b,
                       float* __restrict__ out) {
    int t = threadIdx.x;   // 1024 threads
    __shared__ float gin[640];
    __shared__ float gate[64];
    __shared__ __align__(16) float xg[IMG_FLAT];
    __shared__ __align__(16) float feat[256];
    __shared__ float g[1024];
    __shared__ float z[288];

    // phase 1: pg / ng / curr_rep into gate_in
    if (t < 64) {
        float a = pred_b[t];
        const float* w = pred_w + (size_t)t * 512;
        for (int k = 0; k < 512; ++k) a = fmaf(psum[k], w[k], a);
        gin[512 + t] = a;
    } else if (t < 128) {
        int o = t - 64;
        float a = nred_b[o];
        const float* w = nred_w + (size_t)o * 512;
        for (int k = 0; k < 512; ++k) a = fmaf(nsum[k], w[k], a);
        gin[576 + o] = a;
    } else if (t < 640) {
        gin[t - 128] = akin[t - 128];   // curr_rep
    }
    __syncthreads();

    // phase 2: gate
    if (t < 64) {
        float a = gate_b[t];
        const float* w = gate_w + (size_t)t * 640;
        for (int k = 0; k < 640; ++k) a = fmaf(gin[k], w[k], a);
        gate[t] = sigf(a);
    }
    __syncthreads();

    // phase 3: xg = x_img * gate[channel]
    for (int i = t; i < IMG_FLAT; i += 1024) xg[i] = ximg[i] * gate[i / 36];
    __syncthreads();

    // phase 4: feat = relu(xg @ lin_w^T + b)
    if (t < 256) {
        float a = lin_b[t];
        const float4* w4 = reinterpret_cast<const float4*>(lin_w + (size_t)t * IMG_FLAT);
        const float4* x4 = reinterpret_cast<const float4*>(xg);
        for (int k = 0; k < IMG_FLAT / 4; ++k) {
            float4 wv = w4[k], xv = x4[k];
            a = fmaf(xv.x, wv.x, fmaf(xv.y, wv.y, fmaf(xv.z, wv.z, fmaf(xv.w, wv.w, a))));
        }
        feat[t] = fmaxf(a, 0.0f);
    }
    __syncthreads();

    // phase 5: LSTM preactivations (1024 outputs)
    {
        float a = lstm_bih[t] + lstm_bhh[t];
        const float4* wi = reinterpret_cast<const float4*>(lstm_wih + (size_t)t * H);
        const float4* wh = reinterpret_cast<const float4*>(lstm_whh + (size_t)t * H);
        const float4* f4 = reinterpret_cast<const float4*>(feat);
        const float4* hx4 = reinterpret_cast<const float4*>(hx);
        for (int k = 0; k < H / 4; ++k) {
            float4 wiv = wi[k], whv = wh[k], fv = f4[k], hv = hx4[k];
            a = fmaf(fv.x, wiv.x, fmaf(hv.x, whv.x, a));
            a = fmaf(fv.y, wiv.y, fmaf(hv.y, whv.y, a));
            a = fmaf(fv.z, wiv.z, fmaf(hv.z, whv.z, a));
            a = fmaf(fv.w, wiv.w, fmaf(hv.w, whv.w, a));
        }
        g[t] = a;
    }
    __syncthreads();

    // phase 6: LSTM cell + z assembly
    if (t < 256) {
        float ig = sigf(g[t]);
        float fg = sigf(g[256 + t]);
        float gg = tanhf(g[512 + t]);
        float og = sigf(g[768 + t]);
        float c2 = fg * cx[t] + ig * gg;
        float h2 = og * tanhf(c2);
        out[261 + t] = c2;
        out[5 + t] = h2;
        z[t] = h2;
    } else if (t < 288) {
        z[t] = time_emb[(size_t)tx[0] * E + (t - 256)];
    }
    __syncthreads();

    // phase 7: critic + actor
    if (t < 5) {
        const float* w = (t == 0) ? crit_w : act_w + (size_t)(t - 1) * 288;
        float a = (t == 0) ? crit_b[0] : act_b[t - 1];
        for (int k = 0; k < 288; ++k) a = fmaf(z[k], w[k], a);
        out[t] = a;
    }
}

// ---------------------------------------------------------------------------
extern "C" void kernel_launch(void* const* d_in, const int* in_sizes, int n_in,
                              void* d_out, int out_size, void* d_ws, size_t ws_size,
                              hipStream_t stream) {
    (void)in_sizes; (void)n_in; (void)out_size; (void)ws_size;

    const float* x         = (const float*)d_in[0];
    const float* hx        = (const float*)d_in[1];
    const float* cx        = (const float*)d_in[2];
    const int*   prev_inst = (const int*)d_in[3];
    const int*   curr_inst = (const int*)d_in[4];
    const int*   next_inst = (const int*)d_in[5];
    const int*   tx        = (const int*)d_in[6];
    const float* emb       = (const float*)d_in[7];
    const float* gru_wih   = (const float*)d_in[8];
    const float* gru_whh   = (const float*)d_in[9];
    const float* gru_bih   = (const float*)d_in[10];
    const float* gru_bhh   = (const float*)d_in[11];
    const float* conv1_w   = (const float*)d_in[12];
    const float* conv1_b   = (const float*)d_in[13];
    const float* conv2_w   = (const float*)d_in[14];
    const float* conv2_b   = (const float*)d_in[15];
    const float* conv3_w   = (const float*)d_in[16];
    const float* conv3_b   = (const float*)d_in[17];
    const float* img_w     = (const float*)d_in[18];
    const float* img_b     = (const float*)d_in[19];
    const float* akp_w     = (const float*)d_in[20];
    const float* akp_b     = (const float*)d_in[21];
    const float* akn_w     = (const float*)d_in[22];
    const float* akn_b     = (const float*)d_in[23];
    const float* bil_p_w   = (const float*)d_in[24];
    const float* bil_p_b   = (const float*)d_in[25];
    const float* bil_n_w   = (const float*)d_in[26];
    const float* bil_n_b   = (const float*)d_in[27];
    const float* pred_w    = (const float*)d_in[28];
    const float* pred_b    = (const float*)d_in[29];
    const float* nred_w    = (const float*)d_in[30];
    const float* nred_b    = (const float*)d_in[31];
    const float* gate_w    = (const float*)d_in[32];
    const float* gate_b    = (const float*)d_in[33];
    const float* lin_w     = (const float*)d_in[34];
    const float* lin_b     = (const float*)d_in[35];
    const float* lstm_wih  = (const float*)d_in[36];
    const float* lstm_whh  = (const float*)d_in[37];
    const float* lstm_bih  = (const float*)d_in[38];
    const float* lstm_bhh  = (const float*)d_in[39];
    const float* time_emb  = (const float*)d_in[40];
    const float* crit_w    = (const float*)d_in[41];
    const float* crit_b    = (const float*)d_in[42];
    const float* act_w     = (const float*)d_in[43];
    const float* act_b     = (const float*)d_in[44];

    float* ws    = (float*)d_ws;
    float* x1    = ws + X1_OFF;
    float* x2    = ws + X2_OFF;
    float* ximg  = ws + XIMG_OFF;
    float* e_all = ws + EALL_OFF;
    float* gi    = ws + GI_OFF;
    float* hf    = ws + HF_OFF;
    float* hb    = ws + HB_OFF;
    float* akin  = ws + AKIN_OFF;
    float* psum  = ws + PSUM_OFF;
    float* nsum  = ws + NSUM_OFF;
    float* out   = (float*)d_out;

    // Conv stack (1,3,124,124) -> (128,30,30) -> (64,14,14) -> (64,6,6)
    k_conv_relu<<<(115200 + 255) / 256, 256, 0, stream>>>(x, conv1_w, conv1_b, x1,
                                                          3, 124, 124, 128, 30, 30, 8, 4);
    k_conv_relu<<<(12544 + 255) / 256, 256, 0, stream>>>(x1, conv2_w, conv2_b, x2,
                                                         128, 30, 30, 64, 14, 14, 4, 2);
    k_conv_relu<<<(2304 + 255) / 256, 256, 0, stream>>>(x2, conv3_w, conv3_b, ximg,
                                                        64, 14, 14, 64, 6, 6, 4, 2);

    // Embedding gather (independent of convs)
    k_embed<<<(TOT_ROWS * E + 255) / 256, 256, 0, stream>>>(prev_inst, curr_inst,
                                                            next_inst, emb, e_all);

    // GRU input projection via fp32 WMMA: 1920 waves, 1 M-tile x 4 N-tiles each.
    k_gi_wmma<<<1920, 32, 0, stream>>>(e_all, gru_wih, gru_bih, gi);

    // 6 independent recurrent scans, one workgroup each.
    k_gru<<<6, 768, 0, stream>>>(gi, gru_whh, gru_bhh, hf, hb);

    // curr_rep + image embedding
    k_curr<<<1, 768, 0, stream>>>(hf, hb, ximg, img_w, img_b, akin);

    // prev/next attention
    k_attend<<<2, 512, 0, stream>>>(akin, akp_w, akp_b, akn_w, akn_b,
                                    bil_p_w, bil_p_b, bil_n_w, bil_n_b,
                                    hf, hb, psum, nsum);

    // fused tail
    k_tail<<<1, 1024, 0, stream>>>(akin, psum, nsum, pred_w, pred_b, nred_w, nred_b,
                                   gate_w, gate_b, ximg, lin_w, lin_b,
                                   lstm_wih, lstm_whh, lstm_bih, lstm_bhh,
                                   hx, cx, time_emb, tx, crit_w, crit_b,
                                   act_w, act_b, out);
}